// MASK_39943195853382
// MI455X (gfx1250) — compile-verified
//
#include <hip/hip_runtime.h>
#include <math.h>

// MI455X / gfx1250 wave32 kernel.
// GEMM ([7x64] x [64 x B*F*T]) on the f32 WMMA pipe (V_WMMA_F32_16X16X4_F32).
// All streaming addresses use GVS form (SGPR base via readfirstlane of the
// wave-uniform t, + one 32-bit VGPR offset reused across all 16 k-steps), so
// address arithmetic rides the SALU instead of the VALU. Epilogue is trig-free
// (cos/sin(atan2) -> x/r, y/r via v_rsq_f32; tanh via one v_exp_f32).
// input streamed non-temporally (526 MB, zero reuse); spec/out ride L2.

typedef float v2f __attribute__((ext_vector_type(2)));
typedef float v8f __attribute__((ext_vector_type(8)));

#define KF   5
#define CIN  64
#define TT   1000
#define FF   257
#define BB   8
#define NOUT 7            // K_F + 2
#define T_PER_BLOCK 64    // 8 waves x 8 passes
#define F_PER_BLOCK 32    // two 16-wide WMMA N tiles per wave
#define TF   (TT * FF)    // 257000 elements between consecutive c planes

__device__ __forceinline__ float fast_rcp(float x)  { return __builtin_amdgcn_rcpf(x); }
__device__ __forceinline__ float fast_rsq(float x)  { return __builtin_amdgcn_rsqf(x); }

__global__ __launch_bounds__(256) void mask_df_kernel(
    const float* __restrict__ input,   // (B, C, T, F)
    const float* __restrict__ spec_r,  // (B, 1, F, T)
    const float* __restrict__ spec_i,  // (B, 1, F, T)
    const float* __restrict__ W,       // (7, C)
    const float* __restrict__ bias,    // (7,)
    float* __restrict__ out)           // (B, F, T, 2)
{
    __shared__ float lds_mask[8][32][9];   // stride 9 -> conflict-free (gcd(9,64)=1)
    __shared__ float lds_mag[8][40];       // |spec| for f0-2 .. f0+33
    __shared__ float lds_cp [8][40];       // cos(pha) = sr/|spec|
    __shared__ float lds_sp [8][40];       // sin(pha) = si/|spec|

    const int lane = threadIdx.x & 31;
    const int wave = threadIdx.x >> 5;
    const int b    = blockIdx.z;
    const int f0   = blockIdx.y * F_PER_BLOCK;
    const int t0   = blockIdx.x * T_PER_BLOCK;

    // ---------------- A fragments: W broadcast, resident all kernel ----------
    // A is 16x4 (MxK): lanes 0-15 hold K=0,1 ; lanes 16-31 hold K=2,3.
    const int mrow = lane & 15;
    const int coff = (lane < 16) ? 0 : 2;
    v2f a_frag[16];
#pragma unroll
    for (int k = 0; k < 16; ++k) {
        if (mrow < NOUT) {
            const float* wp = W + mrow * CIN + 4 * k + coff;
            a_frag[k].x = wp[0];
            a_frag[k].y = wp[1];
        } else {
            a_frag[k].x = 0.0f;
            a_frag[k].y = 0.0f;
        }
    }
    float bl[NOUT];
#pragma unroll
    for (int o = 0; o < NOUT; ++o) bl[o] = bias[o];

    // B-frag per-lane f columns (clamped so addresses stay in-bounds; results
    // for overhanging f are masked off at the store).
    const int fA = min(f0 + (lane & 15), FF - 1);
    const int fB = min(f0 + 16 + (lane & 15), FF - 1);
    const int fE = f0 + lane;                 // this lane's epilogue frequency

    // 32-bit divergent offsets, constant across all k-steps / whole kernel.
    const unsigned voffA = (unsigned)(coff * TF + fA);      // <= ~3.1 MB in bytes
    const unsigned voffB = (unsigned)(coff * TF + fB);

    for (int pass = 0; pass < T_PER_BLOCK / 8; ++pass) {
        const int  t      = t0 + pass * 8 + wave;
        const bool tvalid = (t < TT);
        const int  tc     = tvalid ? t : (TT - 1);
        // tc is wave-uniform by construction -> pin it to an SGPR so all
        // global accesses lower to saddr + 32-bit voffset (GVS mode).
        const int  tcs    = __builtin_amdgcn_readfirstlane(tc);

        // ------------- GEMM: mask[o, f0..f0+31] at (b, t) via f32 WMMA -------
        // element (b,c,t,f) = b*C*T*F + c*T*F + t*F + f ; scalar part carries
        // b, tc and k; vector part is voffA/voffB (fixed).
        const float* sb = input + (size_t)b * CIN * TF + (size_t)tcs * FF;
        v8f accA = {0.f,0.f,0.f,0.f,0.f,0.f,0.f,0.f};
        v8f accB = {0.f,0.f,0.f,0.f,0.f,0.f,0.f,0.f};
#pragma unroll
        for (int k = 0; k < 16; ++k) {
            const float* pk  = sb + (size_t)(4 * k) * TF;       // uniform (SGPR)
            const float* pkY = pk + TF;                          // uniform (SGPR)
            v2f bfA, bfB;
            bfA.x = __builtin_nontemporal_load(pk  + voffA);
            bfA.y = __builtin_nontemporal_load(pkY + voffA);
            bfB.x = __builtin_nontemporal_load(pk  + voffB);
            bfB.y = __builtin_nontemporal_load(pkY + voffB);
            accA = __builtin_amdgcn_wmma_f32_16x16x4_f32(
                       false, a_frag[k], false, bfA, (short)0, accA, false, false);
            accB = __builtin_amdgcn_wmma_f32_16x16x4_f32(
                       false, a_frag[k], false, bfB, (short)0, accB, false, false);
        }

        // ------------- stage |spec|, cos(pha), sin(pha) for f0-2..f0+33 ------
        // cos(atan2(y,x)) = x/r ; sin(atan2(y,x)) = y/r  (exact, all quadrants)
        {
            const float* srb = spec_r + (size_t)b * FF * TT + tcs;  // uniform
            const float* sib = spec_i + (size_t)b * FF * TT + tcs;  // uniform
            const int  fm = f0 + lane - 2;
            const bool v  = (fm >= 0) && (fm < FF);
            const unsigned vo = (unsigned)((v ? fm : 0) * TT);      // divergent
            const float sr = v ? srb[vo] : 0.0f;
            const float si = v ? sib[vo] : 0.0f;
            const float q  = sr * sr + si * si;
            const float iv = (q > 0.0f) ? fast_rsq(q) : 0.0f;
            lds_mag[wave][lane] = q * iv;                       // sqrt(q), 0 if q==0
            lds_cp [wave][lane] = (q > 0.0f) ? sr * iv : 1.0f;  // cos(atan2(0,0))=1
            lds_sp [wave][lane] = si * iv;
            if (lane < 4) {
                const int  fm2 = f0 + 30 + lane;
                const bool v2  = (fm2 < FF);
                const unsigned vo2 = (unsigned)((v2 ? fm2 : 0) * TT);
                const float sr2 = v2 ? srb[vo2] : 0.0f;
                const float si2 = v2 ? sib[vo2] : 0.0f;
                const float q2  = sr2 * sr2 + si2 * si2;
                const float iv2 = (q2 > 0.0f) ? fast_rsq(q2) : 0.0f;
                lds_mag[wave][32 + lane] = q2 * iv2;
                lds_cp [wave][32 + lane] = (q2 > 0.0f) ? sr2 * iv2 : 1.0f;
                lds_sp [wave][32 + lane] = si2 * iv2;
            }
        }

        // ------------- spread accumulator to all 32 lanes via LDS ------------
        // acc VGPR r, lanes 0-15: D[M=r][N=lane] = mask[o=r][f-tile pos lane].
        if (lane < 16) {
#pragma unroll
            for (int r = 0; r < NOUT; ++r) {
                lds_mask[wave][lane     ][r] = accA[r];
                lds_mask[wave][lane + 16][r] = accB[r];
            }
        }
        // wave-private LDS region: only need the DS pipe drained, no barrier.
        asm volatile("s_wait_dscnt 0" ::: "memory");

        // ------------- epilogue: one (b, fE, t) output pair per lane ---------
        if (tvalid && fE < FF) {
            float m[NOUT];
#pragma unroll
            for (int r = 0; r < NOUT; ++r)
                m[r] = lds_mask[wave][lane][r] + bl[r];

            // mag_df = sum_k |spec|[f-2+k] * sigmoid(mask_k)
            float mdf = 0.0f;
#pragma unroll
            for (int k = 0; k < KF; ++k) {
                const float s = fast_rcp(1.0f + __expf(-m[k]));
                mdf = fmaf(lds_mag[wave][lane + k], s, mdf);
            }

            // mag_mask = tanh(sqrt(max(s2r^2+s2i^2, 1e-10)))  (arg >= 0)
            const float s2r = m[KF];
            const float s2i = m[KF + 1];
            const float tmg = sqrtf(fmaxf(s2r * s2r + s2i * s2i, 1e-10f));
            const float ee  = __expf(-2.0f * tmg);
            const float mm  = (1.0f - ee) * fast_rcp(1.0f + ee);

            // cos/sin(pha_mask) without atan2
            const float px  = s2r + 1e-10f;
            const float py  = s2i + 1e-10f;
            const float qp  = px * px + py * py;
            const float ivp = (qp > 0.0f) ? fast_rsq(qp) : 0.0f;
            const float cpm = (qp > 0.0f) ? px * ivp : 1.0f;
            const float spm = py * ivp;

            // angle-sum: cos(pha+pm), sin(pha+pm)
            const float cp = lds_cp[wave][lane + 2];
            const float sp = lds_sp[wave][lane + 2];
            const float cr = cp * cpm - sp * spm;
            const float ci = sp * cpm + cp * spm;

            const float amp = mdf * mm;
            // out element (b, fE, t, 0/1): scalar base carries b and t (==tcs),
            // vector offset 2*fE*TT (max ~2 MB in bytes).
            float* ob = out + ((size_t)b * FF * TT + (size_t)tcs) * 2;
            const unsigned vo = (unsigned)fE * (unsigned)(2 * TT);
            ob[vo]     = amp * cr;
            ob[vo + 1] = amp * ci;
        }
    }
}

extern "C" void kernel_launch(void* const* d_in, const int* in_sizes, int n_in,
                              void* d_out, int out_size, void* d_ws, size_t ws_size,
                              hipStream_t stream) {
    const float* input  = (const float*)d_in[0];
    const float* spec_r = (const float*)d_in[1];
    const float* spec_i = (const float*)d_in[2];
    const float* W      = (const float*)d_in[3];
    const float* bias   = (const float*)d_in[4];
    float* out = (float*)d_out;

    dim3 grid((TT + T_PER_BLOCK - 1) / T_PER_BLOCK,   // 16 t-chunks
              (FF + F_PER_BLOCK - 1) / F_PER_BLOCK,   // 9 f-tiles
              BB);                                    // 8 batches
    dim3 block(256);                                  // 8 wave32 waves
    hipLaunchKernelGGL(mask_df_kernel, grid, block, 0, stream,
                       input, spec_r, spec_i, W, bias, out);
}